// DMoN_27831388078434
// MI455X (gfx1250) — compile-verified
//
#include <hip/hip_runtime.h>

typedef __attribute__((ext_vector_type(2))) float v2f;
typedef __attribute__((ext_vector_type(4))) float v4f;
typedef __attribute__((ext_vector_type(8))) float v8f;

#define NN     16384
#define C_IN   128
#define C_HID  64
#define C_OUT  16
#define LDA    132   // 128 + 4 pad: keeps 16B alignment for all rows, no LDS bank conflicts

// ---------------- utility kernels ----------------

__global__ void gcn_zero(float* __restrict__ p, int n) {
    int i = blockIdx.x * blockDim.x + threadIdx.x;
    if (i < n) p[i] = 0.0f;
}

__global__ void gcn_degree(const int* __restrict__ dst, float* __restrict__ deg, int E) {
    int e = blockIdx.x * blockDim.x + threadIdx.x;
    if (e < E) atomicAdd(&deg[dst[e]], 1.0f);
}

__global__ void gcn_invsqrt(const float* __restrict__ deg, float* __restrict__ inv, int n) {
    int i = blockIdx.x * blockDim.x + threadIdx.x;
    if (i < n) inv[i] = rsqrtf(deg[i] + 1.0f);
}

// ---------------- GEMM1: H[N,64] = X[N,128] @ W1[128,64] via fp32 WMMA ----------------
// One block = 4 waves; block owns 16 rows; wave w owns columns [16w, 16w+16).
__global__ __launch_bounds__(128) void gcn_gemm1(const float* __restrict__ X,
                                                 const float* __restrict__ W,
                                                 float* __restrict__ H) {
    __shared__ __align__(16) float As[16 * LDA];
    const int row0 = blockIdx.x * 16;

    // Cooperative coalesced load of the 16x128 A tile (8 threads per row, 16 floats each).
    {
        const int t  = threadIdx.x;
        const int r  = t >> 3;
        const int cb = (t & 7) << 4;
        const float* gp = X + (row0 + r) * C_IN + cb;
        float* lp = As + r * LDA + cb;
        #pragma unroll
        for (int j = 0; j < 4; ++j)
            *(v4f*)(lp + 4 * j) = *(const v4f*)(gp + 4 * j);
    }
    __syncthreads();

    const int lane = threadIdx.x & 31;
    const int wv   = threadIdx.x >> 5;
    const int col0 = wv * 16;
    const int n    = lane & 15;   // A row / B col / D col
    const int hl   = lane >> 4;   // lane half selects K pair

    v8f c = {};
    #pragma unroll
    for (int k0 = 0; k0 < C_IN; k0 += 4) {
        const int kk = k0 + hl * 2;                       // always even -> 8B aligned LDS read
        v2f a = *(const v2f*)(As + n * LDA + kk);         // A: 16x4 fragment
        v2f b;
        b.x = W[kk * C_HID + col0 + n];                   // B: 4x16 fragment
        b.y = W[(kk + 1) * C_HID + col0 + n];
        c = __builtin_amdgcn_wmma_f32_16x16x4_f32(false, a, false, b, (short)0, c, false, false);
    }

    #pragma unroll
    for (int r = 0; r < 8; ++r)
        H[(row0 + hl * 8 + r) * C_HID + col0 + n] = c[r];
}

// ---------------- GEMM2: H2[N,16] = H1[N,64] @ W2[64,16] via fp32 WMMA ----------------
__global__ __launch_bounds__(128) void gcn_gemm2(const float* __restrict__ H1,
                                                 const float* __restrict__ W,
                                                 float* __restrict__ H2) {
    const int lane = threadIdx.x & 31;
    const int wv   = threadIdx.x >> 5;
    const int row0 = (blockIdx.x * 4 + wv) * 16;
    const int n    = lane & 15;
    const int hl   = lane >> 4;

    v8f c = {};
    #pragma unroll
    for (int k0 = 0; k0 < C_HID; k0 += 4) {
        const int kk = k0 + hl * 2;
        v2f a;
        a.x = H1[(row0 + n) * C_HID + kk];
        a.y = H1[(row0 + n) * C_HID + kk + 1];
        v2f b;
        b.x = W[kk * C_OUT + n];
        b.y = W[(kk + 1) * C_OUT + n];
        c = __builtin_amdgcn_wmma_f32_16x16x4_f32(false, a, false, b, (short)0, c, false, false);
    }

    #pragma unroll
    for (int r = 0; r < 8; ++r)
        H2[(row0 + hl * 8 + r) * C_OUT + n] = c[r];
}

// ---------------- edge scatter: agg[dst] += h[src] * inv[src]*inv[dst] ----------------
// 16 threads per edge (layer1, 64 ch) / 4 threads per edge (layer2, 16 ch);
// consecutive lanes cover one source row -> coalesced float4 gathers, broadcast index loads.
__global__ void gcn_scatter64(const int* __restrict__ src, const int* __restrict__ dst,
                              const float* __restrict__ inv, const float* __restrict__ H,
                              float* __restrict__ agg, int E) {
    int tid = blockIdx.x * blockDim.x + threadIdx.x;
    if (tid >= E * 16) return;
    const int e  = tid >> 4;
    const int c4 = (tid & 15) << 2;
    const int s = src[e], d = dst[e];
    const float coef = inv[s] * inv[d];
    const v4f hv = *(const v4f*)(H + s * C_HID + c4);
    float* ap = agg + d * C_HID + c4;
    atomicAdd(ap + 0, hv.x * coef);
    atomicAdd(ap + 1, hv.y * coef);
    atomicAdd(ap + 2, hv.z * coef);
    atomicAdd(ap + 3, hv.w * coef);
}

__global__ void gcn_scatter16(const int* __restrict__ src, const int* __restrict__ dst,
                              const float* __restrict__ inv, const float* __restrict__ H,
                              float* __restrict__ agg, int E) {
    int tid = blockIdx.x * blockDim.x + threadIdx.x;
    if (tid >= E * 4) return;
    const int e  = tid >> 2;
    const int c4 = (tid & 3) << 2;
    const int s = src[e], d = dst[e];
    const float coef = inv[s] * inv[d];
    const v4f hv = *(const v4f*)(H + s * C_OUT + c4);
    float* ap = agg + d * C_OUT + c4;
    atomicAdd(ap + 0, hv.x * coef);
    atomicAdd(ap + 1, hv.y * coef);
    atomicAdd(ap + 2, hv.z * coef);
    atomicAdd(ap + 3, hv.w * coef);
}

// ---------------- self-loop + bias + relu (layer 1) ----------------
__global__ void gcn_self_relu(const float* __restrict__ agg, const float* __restrict__ H,
                              const float* __restrict__ inv, const float* __restrict__ b,
                              float* __restrict__ H1, int total) {
    int tid = blockIdx.x * blockDim.x + threadIdx.x;
    if (tid >= total) return;
    const int i = tid >> 6;     // /64
    const int c = tid & 63;
    const float s = inv[i] * inv[i];
    const float v = agg[tid] + H[tid] * s + b[c];
    H1[tid] = v > 0.0f ? v : 0.0f;
}

// ---------------- self-loop + bias + row softmax (layer 2) ----------------
__global__ void gcn_softmax(const float* __restrict__ agg2, const float* __restrict__ H2,
                            const float* __restrict__ inv, const float* __restrict__ b,
                            float* __restrict__ out, int n) {
    int r = blockIdx.x * blockDim.x + threadIdx.x;
    if (r >= n) return;
    const float s = inv[r] * inv[r];
    float l[C_OUT];
    float mx = -1e30f;
    #pragma unroll
    for (int c = 0; c < C_OUT; ++c) {
        l[c] = agg2[r * C_OUT + c] + H2[r * C_OUT + c] * s + b[c];
        mx = fmaxf(mx, l[c]);
    }
    float sum = 0.0f;
    #pragma unroll
    for (int c = 0; c < C_OUT; ++c) { l[c] = __expf(l[c] - mx); sum += l[c]; }
    const float rs = 1.0f / sum;
    #pragma unroll
    for (int c = 0; c < C_OUT; ++c) out[r * C_OUT + c] = l[c] * rs;
}

// ---------------- launch ----------------
extern "C" void kernel_launch(void* const* d_in, const int* in_sizes, int n_in,
                              void* d_out, int out_size, void* d_ws, size_t ws_size,
                              hipStream_t stream) {
    const float* x  = (const float*)d_in[0];
    const int*   ei = (const int*)  d_in[1];   // int32 per harness convention
    const float* W1 = (const float*)d_in[3];
    const float* b1 = (const float*)d_in[4];
    const float* W2 = (const float*)d_in[5];
    const float* b2 = (const float*)d_in[6];
    float* out = (float*)d_out;

    const int N = NN;
    const int E = in_sizes[1] / 2;
    const int* src = ei;
    const int* dst = ei + E;

    // workspace layout (floats): [deg N][agg1 N*64][agg2 N*16][inv N][h N*64][h1 N*64][h2 N*16]
    float* ws   = (float*)d_ws;
    float* deg  = ws;
    float* agg1 = deg  + N;
    float* agg2 = agg1 + N * C_HID;
    float* inv  = agg2 + N * C_OUT;
    float* h    = inv  + N;
    float* h1   = h    + N * C_HID;
    float* h2   = h1   + N * C_HID;

    const int zn = N * (1 + C_HID + C_OUT);   // deg, agg1, agg2 are contiguous
    gcn_zero   <<<(zn + 255) / 256, 256, 0, stream>>>(deg, zn);
    gcn_degree <<<(E + 255) / 256, 256, 0, stream>>>(dst, deg, E);
    gcn_invsqrt<<<(N + 255) / 256, 256, 0, stream>>>(deg, inv, N);

    gcn_gemm1  <<<N / 16, 128, 0, stream>>>(x, W1, h);
    gcn_scatter64<<<(E * 16 + 255) / 256, 256, 0, stream>>>(src, dst, inv, h, agg1, E);
    gcn_self_relu<<<(N * C_HID + 255) / 256, 256, 0, stream>>>(agg1, h, inv, b1, h1, N * C_HID);

    gcn_gemm2  <<<N / 64, 128, 0, stream>>>(h1, W2, h2);
    gcn_scatter16<<<(E * 4 + 255) / 256, 256, 0, stream>>>(src, dst, inv, h2, agg2, E);
    gcn_softmax<<<(N + 255) / 256, 256, 0, stream>>>(agg2, h2, inv, b2, out, N);
}